// AdditiveAttention_15831249453729
// MI455X (gfx1250) — compile-verified
//
#include <hip/hip_runtime.h>
#include <hip/hip_bf16.h>

typedef __attribute__((ext_vector_type(2))) float v2f;
typedef __attribute__((ext_vector_type(8))) float v8f;

#define NB   16   // batch
#define NLQ  256  // query length
#define NLK  256  // key length
#define ND   256  // input feature dim (DQ=DK=DV)
#define NH   256  // hidden dim

// gfx1250 native tanh (TRANS op). Builtin preferred; raw-ISA fallback with a
// nop to respect the 1-cycle trans-result hazard if the builtin is absent.
__device__ __forceinline__ float aa_tanh(float x) {
#if __has_builtin(__builtin_amdgcn_tanhf)
  return __builtin_amdgcn_tanhf(x);
#else
  float r;
  asm volatile("v_tanh_f32 %0, %1\n\ts_nop 1" : "=v"(r) : "v"(x));
  return r;
#endif
}

// ---------------------------------------------------------------------------
// K1: Q/K projections via V_WMMA_F32_16X16X4_F32.
//   blockIdx.y == 0 : qproj[b*LQ+q][h]      = (queries @ W_q)
//   blockIdx.y == 1 : kprojT[b][h][k]       = (keys    @ W_k) transposed store
// One 16x16 output tile per wave, K stepped by 4 (f32 WMMA shape).
// A frag (16x4): m = lane%16, K = 2*(lane/16) + vgpr
// B frag (4x16): n = lane%16, K = 2*(lane/16) + vgpr
// D (16x16):     vgpr r -> M = r + 8*(lane/16), N = lane%16
// ---------------------------------------------------------------------------
__global__ __launch_bounds__(128) void aa_proj_kernel(
    const float* __restrict__ queries, const float* __restrict__ keys,
    const float* __restrict__ Wq, const float* __restrict__ Wk,
    float* __restrict__ qproj, float* __restrict__ kprojT) {
  const int wave = threadIdx.x >> 5;
  const int lane = threadIdx.x & 31;
  const int tile = blockIdx.x * 4 + wave;   // 4096 tiles: 256 row-tiles x 16 col-tiles
  const int row0 = (tile >> 4) << 4;        // over B*LQ = 4096 rows
  const int col0 = (tile & 15) << 4;        // over H = 256 cols
  const bool isK = (blockIdx.y != 0);

  const float* X = isK ? keys : queries;    // (4096, 256) row-major
  const float* W = isK ? Wk : Wq;           // (256, 256) row-major

  const int hl = lane >> 4;                 // lane half: selects K pair
  const int mr = lane & 15;
  const int koff = 2 * hl;

  const float* Arow = X + (size_t)(row0 + mr) * ND;
  const float* Bcol = W + col0 + mr;

  v8f c = {};
#pragma unroll 8
  for (int k = 0; k < ND; k += 4) {
    v2f a, b;
    a.x = Arow[k + koff];
    a.y = Arow[k + koff + 1];
    b.x = Bcol[(size_t)(k + koff) * NH];
    b.y = Bcol[(size_t)(k + koff + 1) * NH];
    c = __builtin_amdgcn_wmma_f32_16x16x4_f32(false, a, false, b, (short)0, c,
                                              false, false);
  }

  if (!isK) {
#pragma unroll
    for (int r = 0; r < 8; ++r) {
      const int m = r + 8 * hl;
      qproj[(size_t)(row0 + m) * NH + col0 + mr] = c[r];
    }
  } else {
    const int bb  = row0 >> 8;    // batch (row tiles never cross batch: 256%16==0)
    const int kkb = row0 & 255;   // key index base within batch
#pragma unroll
    for (int r = 0; r < 8; ++r) {
      const int m = r + 8 * hl;   // key index offset
      kprojT[(size_t)bb * NH * NLK + (size_t)(col0 + mr) * NLK + kkb + m] = c[r];
    }
  }
}

// ---------------------------------------------------------------------------
// K2: additive scores + masked softmax. One block per (b, q) row; thread t
// owns key column k=t. 256 v_tanh_f32 per thread -> 268M total (the hot loop).
// ---------------------------------------------------------------------------
__global__ __launch_bounds__(256) void aa_scores_softmax_kernel(
    const float* __restrict__ qproj, const float* __restrict__ kprojT,
    const int* __restrict__ valid_lens, const float* __restrict__ w_v,
    float* __restrict__ attn) {
  __shared__ float sh_q[NH];
  __shared__ float sh_w[NH];
  __shared__ float red[256];

  const int b = blockIdx.y;
  const int q = blockIdx.x;
  const int t = threadIdx.x;  // key index

  sh_q[t] = qproj[(size_t)(b * NLQ + q) * NH + t];
  sh_w[t] = w_v[t];
  __syncthreads();

  const float* kcol = kprojT + (size_t)b * NH * NLK + t;  // stride NLK over h
  float s = 0.f;
#pragma unroll 8
  for (int h = 0; h < NH; ++h) {
    s = fmaf(sh_w[h], aa_tanh(sh_q[h] + kcol[(size_t)h * NLK]), s);
  }

  const int vlen = valid_lens[b];
  const float sc = (t < vlen) ? s : -1.0e6f;

  // row max
  red[t] = sc;
  __syncthreads();
  for (int off = 128; off > 0; off >>= 1) {
    if (t < off) red[t] = fmaxf(red[t], red[t + off]);
    __syncthreads();
  }
  const float mx = red[0];
  __syncthreads();

  const float e = __expf(sc - mx);

  // row sum
  red[t] = e;
  __syncthreads();
  for (int off = 128; off > 0; off >>= 1) {
    if (t < off) red[t] += red[t + off];
    __syncthreads();
  }
  const float inv = 1.f / red[0];

  attn[(size_t)(b * NLQ + q) * NLK + t] = e * inv;
}

// ---------------------------------------------------------------------------
// K3: out = attn @ values, per-batch 256x256x256 GEMM via f32 WMMA.
// ---------------------------------------------------------------------------
__global__ __launch_bounds__(128) void aa_av_kernel(
    const float* __restrict__ attn, const float* __restrict__ values,
    float* __restrict__ out) {
  const int wave = threadIdx.x >> 5;
  const int lane = threadIdx.x & 31;
  const int tile = blockIdx.x * 4 + wave;   // 4096 tiles: b(16) x rowT(16) x colT(16)
  const int b    = tile >> 8;
  const int row0 = ((tile >> 4) & 15) << 4;
  const int col0 = (tile & 15) << 4;

  const float* A = attn + (size_t)b * NLQ * NLK;
  const float* V = values + (size_t)b * NLK * ND;

  const int hl = lane >> 4;
  const int mr = lane & 15;
  const int koff = 2 * hl;

  const float* Arow = A + (size_t)(row0 + mr) * NLK;
  const float* Bcol = V + col0 + mr;

  v8f c = {};
#pragma unroll 8
  for (int k = 0; k < NLK; k += 4) {
    v2f a, bv;
    a.x = Arow[k + koff];
    a.y = Arow[k + koff + 1];
    bv.x = Bcol[(size_t)(k + koff) * ND];
    bv.y = Bcol[(size_t)(k + koff + 1) * ND];
    c = __builtin_amdgcn_wmma_f32_16x16x4_f32(false, a, false, bv, (short)0, c,
                                              false, false);
  }

  float* O = out + (size_t)b * NLQ * ND;
#pragma unroll
  for (int r = 0; r < 8; ++r) {
    const int m = r + 8 * hl;
    O[(size_t)(row0 + m) * ND + col0 + mr] = c[r];
  }
}

extern "C" void kernel_launch(void* const* d_in, const int* in_sizes, int n_in,
                              void* d_out, int out_size, void* d_ws,
                              size_t ws_size, hipStream_t stream) {
  const float* queries    = (const float*)d_in[0];
  const float* keys       = (const float*)d_in[1];
  const float* values     = (const float*)d_in[2];
  const int*   valid_lens = (const int*)d_in[3];
  const float* Wq         = (const float*)d_in[4];
  const float* Wk         = (const float*)d_in[5];
  const float* wv         = (const float*)d_in[6];
  float* out = (float*)d_out;

  // Workspace: qproj (4MB) | kprojT (4MB) | attn (4MB)
  float* qproj  = (float*)d_ws;
  float* kprojT = qproj + (size_t)NB * NLQ * NH;
  float* attn   = kprojT + (size_t)NB * NH * NLK;

  // K1: both projections (y=0 -> Q, y=1 -> K); 4096 tiles / 4 waves per block.
  dim3 gproj(1024, 2);
  aa_proj_kernel<<<gproj, 128, 0, stream>>>(queries, keys, Wq, Wk, qproj,
                                            kprojT);

  // K2: one block per (b, q) row.
  dim3 gs(NLQ, NB);
  aa_scores_softmax_kernel<<<gs, 256, 0, stream>>>(qproj, kprojT, valid_lens,
                                                   wv, attn);

  // K3: attn @ values.
  aa_av_kernel<<<1024, 128, 0, stream>>>(attn, values, out);
}